// AdaLoRAWithBase_7825430413636
// MI455X (gfx1250) — compile-verified
//
#include <hip/hip_runtime.h>
#include <hip/hip_bf16.h>
#include <math.h>

// ---------------------------------------------------------------------------
// AdaLoRAWithBase, factored form:
//   h  = gelu(ada @ w1 + b1)                [16,1024]
//   xw = h @ w2 + b2                        [16,32768]  (x_a | x_b halves)
//   t[b,r]   = sum_c x[b,c] * xw[b, c*8+r]            [16,8]
//   out[b,o] = x[b,o] + sum_c x[b,c]*base[c,o]
//            + sum_r t[b,r] * xw[b, 16384 + o*8 + r]
// All GEMMs have M=16 == one v_wmma_f32_16x16x4_f32 tile row.
// Memory-bound: w2 (128MB) + base (16MB) + w1 (4MB) ~ 6.4us at 23.3 TB/s.
// ---------------------------------------------------------------------------

typedef __attribute__((ext_vector_type(2))) float v2f;
typedef __attribute__((ext_vector_type(8))) float v8f;

__device__ __forceinline__ v8f wmma_f32_k4(v2f a, v2f b, v8f c) {
  // D = A(16x4,f32) * B(4x16,f32) + C(16x16,f32)
  return __builtin_amdgcn_wmma_f32_16x16x4_f32(
      /*neg_a=*/false, a, /*neg_b=*/false, b,
      /*c_mod=*/(short)0, c, /*reuse_a=*/false, /*reuse_b=*/false);
}

// MODE: 0 = out = acc + bias
//       1 = out = gelu(acc + bias)     (exact, erf-based)
//       2 = out = resid + acc + sum_r t[m,r]*xw[m,16384+n*8+r]
template <int K, int N, int NT, int MODE>
__global__ __launch_bounds__(256) void skinny_gemm(
    const float* __restrict__ A,     // [16, K]
    const float* __restrict__ W,     // [K, N] row-major
    const float* __restrict__ bias,  // [N]           (MODE 0/1)
    float* __restrict__ out,         // [16, N]
    const float* __restrict__ resid, // [16, N]       (MODE 2)
    const float* __restrict__ tvec,  // [16, 8]       (MODE 2)
    const float* __restrict__ xw)    // [16, 32768]   (MODE 2)
{
  const int tid   = threadIdx.x;
  const int wave  = tid >> 5;       // 8 waves split K
  const int lane  = tid & 31;
  const int l16   = lane & 15;
  const int khalf = lane >> 4;      // 0: K pair {0,1}, 1: K pair {2,3}
  const int n0    = blockIdx.x * (16 * NT);

  const int KPW  = K / 8;           // K-range per wave
  const int kbeg = wave * KPW;

  // Per-lane bases. A fp32 fragment (16x4): lane m = l16, vgpr j -> k = khalf*2+j
  const float* Arow = A + (size_t)l16 * K;
  // B fp32 fragment (4x16): lane n = l16, vgpr j -> k = khalf*2+j
  const float* Wcol = W + n0 + l16;

  v8f acc[NT];
#pragma unroll
  for (int t = 0; t < NT; ++t) acc[t] = v8f{};

  for (int kk = kbeg; kk < kbeg + KPW; kk += 4) {
    const int ka = kk + khalf * 2;
    const v2f a = *(const v2f*)(Arow + ka);           // 8B-aligned (ka even)
    // Unconditional clamped prefetch ~16 rows ahead on the weight stream:
    // single VALU min instead of a divergent branch (no EXEC churn).
    const int kp = min(ka + 16, K - 1);
    __builtin_prefetch(Wcol + (size_t)kp * N, 0, 1);
#pragma unroll
    for (int t = 0; t < NT; ++t) {
      v2f b;
      b.x = Wcol[(size_t)ka * N + t * 16];
      b.y = Wcol[(size_t)(ka + 1) * N + t * 16];
      acc[t] = wmma_f32_k4(a, b, acc[t]);
    }
  }

  // Cross-wave K reduction through LDS.
  __shared__ float red[8 * NT * 256];
#pragma unroll
  for (int t = 0; t < NT; ++t)
#pragma unroll
    for (int i = 0; i < 8; ++i)
      red[(wave * NT + t) * 256 + i * 32 + lane] = acc[t][i];
  __syncthreads();

  // Thread tid owns C/D slot (vgpr i = tid>>5, lane L = tid&31):
  //   m = (L>>4)*8 + i, n = n0 + t*16 + (L&15)
  const int i = tid >> 5;
  const int L = tid & 31;
  const int m = (L >> 4) * 8 + i;
#pragma unroll
  for (int t = 0; t < NT; ++t) {
    float s = 0.f;
#pragma unroll
    for (int w = 0; w < 8; ++w) s += red[(w * NT + t) * 256 + tid];
    const int n = n0 + t * 16 + (L & 15);
    if constexpr (MODE == 0) {
      out[(size_t)m * N + n] = s + bias[n];
    } else if constexpr (MODE == 1) {
      const float v = s + bias[n];
      out[(size_t)m * N + n] = 0.5f * v * (1.0f + erff(v * 0.70710678118654752f));
    } else {
      float corr = 0.f;
      const float* tb  = tvec + m * 8;
      const float* xbp = xw + (size_t)m * 32768 + 16384 + (size_t)n * 8;
#pragma unroll
      for (int r = 0; r < 8; ++r) corr += tb[r] * xbp[r];
      out[(size_t)m * N + n] = resid[(size_t)m * N + n] + s + corr;
    }
  }
}

// t[b,r] = sum_c x[b,c] * xw[b, c*8 + r]   (x_a half of xw)
__global__ __launch_bounds__(256) void rank_reduce(
    const float* __restrict__ x,   // [16, 2048]
    const float* __restrict__ xw,  // [16, 32768]
    float* __restrict__ t)         // [16, 8]
{
  const int b   = blockIdx.x;
  const int tid = threadIdx.x;
  const int r   = tid & 7;
  const int c0  = tid >> 3;  // 0..31
  const float* xr  = x + (size_t)b * 2048;
  const float* xar = xw + (size_t)b * 32768;
  float s = 0.f;
  for (int c = c0; c < 2048; c += 32) s += xr[c] * xar[(size_t)c * 8 + r];
  __shared__ float red[256];
  red[tid] = s;
  __syncthreads();
  if (tid < 8) {
    float tot = 0.f;
#pragma unroll
    for (int j = 0; j < 32; ++j) tot += red[tid + j * 8];
    t[b * 8 + tid] = tot;
  }
}

extern "C" void kernel_launch(void* const* d_in, const int* in_sizes, int n_in,
                              void* d_out, int out_size, void* d_ws, size_t ws_size,
                              hipStream_t stream) {
  const float* x    = (const float*)d_in[0];  // [16,2048]
  const float* ada  = (const float*)d_in[1];  // [16,1024]
  const float* base = (const float*)d_in[2];  // [2048,2048]
  const float* w1   = (const float*)d_in[3];  // [1024,1024]
  const float* b1   = (const float*)d_in[4];  // [1024]
  const float* w2   = (const float*)d_in[5];  // [1024,32768]
  const float* b2   = (const float*)d_in[6];  // [32768]
  float* out = (float*)d_out;                 // [16,2048]

  float* h  = (float*)d_ws;                   // 16*1024
  float* xw = h + 16 * 1024;                  // 16*32768
  float* tv = xw + 16 * 32768;                // 16*8

  // h = gelu(ada @ w1 + b1): N=1024 -> 64 tiles, NT=1
  skinny_gemm<1024, 1024, 1, 1><<<64, 256, 0, stream>>>(
      ada, w1, b1, h, nullptr, nullptr, nullptr);

  // xw = h @ w2 + b2: N=32768 -> NT=4, 512 blocks (dominant: streams 128MB of w2)
  skinny_gemm<1024, 32768, 4, 0><<<512, 256, 0, stream>>>(
      h, w2, b2, xw, nullptr, nullptr, nullptr);

  // t[b,r] = x . x_a
  rank_reduce<<<16, 256, 0, stream>>>(x, xw, tv);

  // out = x + x @ base + t . x_b^T
  skinny_gemm<2048, 2048, 1, 2><<<128, 256, 0, stream>>>(
      x, base, nullptr, out, x, tv, xw);
}